// ChamferDistanceLoss_81698867904729
// MI455X (gfx1250) — compile-verified
//
#include <hip/hip_runtime.h>

typedef __attribute__((ext_vector_type(2))) float v2f;
typedef __attribute__((ext_vector_type(8))) float v8f;

#define B_   32
#define N_   16384
#define S_   2048
#define TILES (S_ / 16)   // 128 column tiles of 16

// ---------------------------------------------------------------------------
// Kernel 0: reset workspace (col-min bits to +inf, row accumulator to 0)
// ---------------------------------------------------------------------------
__global__ void chamfer_init(unsigned int* __restrict__ colmin_bits,
                             float* __restrict__ rowAccum, int n) {
    int i = blockIdx.x * blockDim.x + threadIdx.x;
    if (i < n) colmin_bits[i] = 0x7F800000u;     // +inf bit pattern
    if (i == 0) rowAccum[0] = 0.0f;
}

// ---------------------------------------------------------------------------
// Kernel 1: main chamfer tiles. One wave32 block per (batch, row-tile).
// dist2(i,j) = n1[i] + n2[j] - 2*dot(p1_i, p2_j); dot via WMMA F32 16x16x4.
// ---------------------------------------------------------------------------
__global__ __launch_bounds__(32) void chamfer_main(
    const float* __restrict__ points1, const float* __restrict__ points2,
    const int*   __restrict__ idx1,    const int*   __restrict__ idx2,
    unsigned int* __restrict__ colmin_bits, float* __restrict__ rowAccum)
{
    __shared__ float sP1[16][3];
    __shared__ float sN1[16];
    __shared__ float sP2[16][3];
    __shared__ float sN2[16];

    const int lane = threadIdx.x;
    const int b    = blockIdx.x / TILES;
    const int rt   = blockIdx.x % TILES;
    const int m    = lane & 15;
    const bool hi  = (lane >= 16);

    // ---- gather 16 p1 rows for this tile (+ squared norms) into LDS ----
    if (lane < 16) {
        int r  = rt * 16 + lane;
        int id = idx1[r];
        const float* p = points1 + ((size_t)b * N_ + (size_t)id) * 3;
        float x = p[0], y = p[1], z = p[2];
        sP1[lane][0] = x; sP1[lane][1] = y; sP1[lane][2] = z;
        sN1[lane] = x * x + y * y + z * z;
    }
    __syncthreads();

    // A operand (16x4 f32, M=lane&15): lanes 0-15 carry K=0,1 ; lanes 16-31 K=2,3
    v2f a;
    a.x = hi ? sP1[m][2] : sP1[m][0];
    a.y = hi ? 0.0f      : sP1[m][1];
    const float n1_lo = sN1[m];   // row norms needed in epilogue
    // full row-norm table stays in LDS (sN1), read per-k below

    float rowMin[8];
#pragma unroll
    for (int k = 0; k < 8; ++k) rowMin[k] = __builtin_inff();

    for (int ct = 0; ct < TILES; ++ct) {
        __syncthreads();   // protect sP2/sN2 reuse across iterations
        if (lane < 16) {
            int cidx = ct * 16 + lane;
            int id   = idx2[cidx];
            const float* p = points2 + ((size_t)b * N_ + (size_t)id) * 3;
            float x = p[0], y = p[1], z = p[2];
            sP2[lane][0] = x; sP2[lane][1] = y; sP2[lane][2] = z;
            sN2[lane] = x * x + y * y + z * z;
        }
        __syncthreads();

        // B operand (4x16 f32, N=lane&15): lanes 0-15 K=0,1 ; lanes 16-31 K=2,3
        v2f bv;
        bv.x = hi ? sP2[m][2] : sP2[m][0];
        bv.y = hi ? 0.0f      : sP2[m][1];

        v8f c = {};
        // (neg_a, A, neg_b, B, c_mod, C, reuse_a, reuse_b)
        c = __builtin_amdgcn_wmma_f32_16x16x4_f32(
                false, a, false, bv, (short)0, c, false, false);

        const float n2 = sN2[m];
        float cmin = __builtin_inff();
#pragma unroll
        for (int k = 0; k < 8; ++k) {
            int M = k + (hi ? 8 : 0);
            float d2 = sN1[M] + n2 - 2.0f * c[k];
            d2 = fmaxf(d2, 0.0f);
            rowMin[k] = fminf(rowMin[k], d2);
            cmin = fminf(cmin, d2);
        }
        // lane l and lane l^16 hold the same column (different row halves)
        cmin = fminf(cmin, __shfl_xor(cmin, 16, 32));
        if (lane < 16) {
            atomicMin(&colmin_bits[(size_t)b * S_ + (size_t)(ct * 16 + lane)],
                      __float_as_uint(cmin));
        }
    }

    // ---- reduce row mins across the 16 lanes of each half, then accumulate ----
    float rsum = 0.0f;
#pragma unroll
    for (int k = 0; k < 8; ++k) {
        float v = rowMin[k];
        v = fminf(v, __shfl_xor(v, 1, 32));
        v = fminf(v, __shfl_xor(v, 2, 32));
        v = fminf(v, __shfl_xor(v, 4, 32));
        v = fminf(v, __shfl_xor(v, 8, 32));
        rsum += __builtin_sqrtf(v);   // lanes 0..15: row k ; lanes 16..31: row k+8
    }
    (void)n1_lo;
    if (lane == 0 || lane == 16) atomicAdd(rowAccum, rsum);
}

// ---------------------------------------------------------------------------
// Kernel 2: final reduction. sqrt-sum all column mins, add row accumulator,
// divide by B*S (W1 = W2 = 1 collapses to a single global mean).
// ---------------------------------------------------------------------------
__global__ __launch_bounds__(256) void chamfer_finish(
    const unsigned int* __restrict__ colmin_bits,
    const float* __restrict__ rowAccum, float* __restrict__ out)
{
    __shared__ float red[256];
    float s = 0.0f;
    for (int i = threadIdx.x; i < B_ * S_; i += 256)
        s += __builtin_sqrtf(__uint_as_float(colmin_bits[i]));
    red[threadIdx.x] = s;
    __syncthreads();
    for (int off = 128; off > 0; off >>= 1) {
        if (threadIdx.x < off) red[threadIdx.x] += red[threadIdx.x + off];
        __syncthreads();
    }
    if (threadIdx.x == 0)
        out[0] = (red[0] + rowAccum[0]) / (float)(B_ * S_);
}

// ---------------------------------------------------------------------------
extern "C" void kernel_launch(void* const* d_in, const int* in_sizes, int n_in,
                              void* d_out, int out_size, void* d_ws, size_t ws_size,
                              hipStream_t stream) {
    const float* points1 = (const float*)d_in[0];
    const float* points2 = (const float*)d_in[1];
    const int*   idx1    = (const int*)d_in[2];
    const int*   idx2    = (const int*)d_in[3];
    float*       out     = (float*)d_out;

    unsigned int* colmin_bits = (unsigned int*)d_ws;            // B_*S_ u32
    float*        rowAccum    = (float*)(colmin_bits + B_ * S_); // 1 float

    const int nInit = B_ * S_;
    chamfer_init<<<(nInit + 255) / 256, 256, 0, stream>>>(colmin_bits, rowAccum, nInit);

    chamfer_main<<<B_ * TILES, 32, 0, stream>>>(
        points1, points2, idx1, idx2, colmin_bits, rowAccum);

    chamfer_finish<<<1, 256, 0, stream>>>(colmin_bits, rowAccum, out);
}